// DisenGCN_63728724738085
// MI455X (gfx1250) — compile-verified
//
#include <hip/hip_runtime.h>
#include <hip/hip_bf16.h>
#include <math.h>

#define N_NODES 50000
#define N_EDGES 800000
#define DIM     128

typedef float v2f __attribute__((ext_vector_type(2)));
typedef float v8f __attribute__((ext_vector_type(8)));

// ---------------------------------------------------------------------------
// C[M,128] = A[M,128] @ W[128,128] + bias   via V_WMMA_F32_16X16X4_F32
// One 16x16 C tile per wave; 8 waves per block cover all 128 output columns
// for one 16-row stripe.  M = 50000 = 3125 * 16 (no tail).
// ---------------------------------------------------------------------------
__global__ void __launch_bounds__(256) gemm_bias_wmma(
    const float* __restrict__ A, const float* __restrict__ W,
    const float* __restrict__ bias, float* __restrict__ C) {
  const int lane = threadIdx.x & 31;
  const int wave = threadIdx.x >> 5;            // N-tile index 0..7
  const int row0 = blockIdx.x << 4;             // 16 rows per block
  const int n    = (wave << 4) + (lane & 15);   // output column
  const int half = lane >> 4;                   // 0: K+0/1, 1: K+2/3
  const int m    = row0 + (lane & 15);          // A row for this lane
  const float* arow = A + (size_t)m * DIM;

  v8f acc = {};
  #pragma unroll 4
  for (int k0 = 0; k0 < DIM; k0 += 4) {
    const int ka = k0 + (half << 1);
    v2f a, b;
    a.x = arow[ka];
    a.y = arow[ka + 1];
    b.x = W[ka * DIM + n];
    b.y = W[(ka + 1) * DIM + n];
    acc = __builtin_amdgcn_wmma_f32_16x16x4_f32(false, a, false, b,
                                                (short)0, acc, false, false);
  }
  const float bs = bias[n];
  float* crow = C + (size_t)(row0 + (half << 3)) * DIM + n;
  #pragma unroll
  for (int r = 0; r < 8; ++r)
    crow[(size_t)r * DIM] = acc[r] + bs;        // row M = row0 + 8*half + r
}

// ---------------------------------------------------------------------------
// Per-channel L2 normalize: 4 channels of 32.  128 threads/block = 4 wave32s,
// one wave per channel; intra-wave xor-shuffle reduction.  in==out is safe.
// ---------------------------------------------------------------------------
__global__ void __launch_bounds__(128) channel_norm(
    const float* __restrict__ in, float* __restrict__ out) {
  const int node = blockIdx.x;
  const int t    = threadIdx.x;
  const size_t idx = (size_t)node * DIM + t;
  const float v = in[idx];
  float ss = v * v;
  #pragma unroll
  for (int off = 16; off; off >>= 1) ss += __shfl_xor(ss, off, 32);
  const float nrm = sqrtf(ss);
  out[idx] = v / fmaxf(nrm, 1e-12f);
}

// ---------------------------------------------------------------------------
// One routing iteration over the edges.  One wave32 per edge:
//   lane l holds elements [4l, 4l+3] of z = xn[src] and cg = c[trg] (float4).
//   s[l]   = sum of lane's 4 cg elements (groups-of-4 reduce is per-lane!)
//   p[k]   = sum_a z[32k+a] * s[a]   -> 4 broadcasts + dot4 + 8-lane reduce
//   softmax over the 4 channels via xor-shuffles at 8/16/24
//   atomicAdd(w * z) into acc[trg]  (L2-resident segment_sum)
// ---------------------------------------------------------------------------
__global__ void __launch_bounds__(256) route_edges(
    const float* __restrict__ xn, const float* __restrict__ c,
    float* __restrict__ acc,
    const int* __restrict__ src, const int* __restrict__ trg) {
  const int e = blockIdx.x * (blockDim.x >> 5) + (threadIdx.x >> 5);
  if (e >= N_EDGES) return;
  const int lane = threadIdx.x & 31;
  const int s = src[e];
  const int t = trg[e];

  const float4 z  = reinterpret_cast<const float4*>(xn + (size_t)s * DIM)[lane];
  const float4 cg = reinterpret_cast<const float4*>(c  + (size_t)t * DIM)[lane];

  // s[a] for a == lane
  const float sv = cg.x + cg.y + cg.z + cg.w;

  // lane l covers channel k = l>>3, positions a = 4*(l&7)+j
  const int base = (lane & 7) << 2;
  const float s0 = __shfl(sv, base + 0);
  const float s1 = __shfl(sv, base + 1);
  const float s2 = __shfl(sv, base + 2);
  const float s3 = __shfl(sv, base + 3);
  float p = z.x * s0 + z.y * s1 + z.z * s2 + z.w * s3;
  p += __shfl_xor(p, 1);
  p += __shfl_xor(p, 2);
  p += __shfl_xor(p, 4);           // p = logits[own channel], replicated x8

  const float p0 = p;
  const float p1 = __shfl_xor(p, 8);
  const float p2 = __shfl_xor(p, 16);
  const float p3 = __shfl_xor(p, 24);
  const float mx = fmaxf(fmaxf(p0, p1), fmaxf(p2, p3));
  const float e0 = expf(p0 - mx), e1 = expf(p1 - mx);
  const float e2 = expf(p2 - mx), e3 = expf(p3 - mx);
  const float w  = e0 / (e0 + e1 + e2 + e3);   // softmax weight of own channel

  float* dst = acc + (size_t)t * DIM + (lane << 2);
  atomicAdd(dst + 0, w * z.x);
  atomicAdd(dst + 1, w * z.y);
  atomicAdd(dst + 2, w * z.z);
  atomicAdd(dst + 3, w * z.w);
}

// ---------------------------------------------------------------------------
extern "C" void kernel_launch(void* const* d_in, const int* in_sizes, int n_in,
                              void* d_out, int out_size, void* d_ws, size_t ws_size,
                              hipStream_t stream) {
  const float* feat  = (const float*)d_in[0];
  const int*   eidx  = (const int*)d_in[1];       // [2, 800000] flattened
  const float* lin_w = (const float*)d_in[2];
  const float* lin_b = (const float*)d_in[3];
  const float* mlp_w = (const float*)d_in[4];
  const float* mlp_b = (const float*)d_in[5];

  const int* src = eidx;
  const int* trg = eidx + N_EDGES;

  const size_t NB = (size_t)N_NODES * DIM;        // floats per node buffer
  float* out_buf = (float*)d_out;                 // [50000,128] out
  float* x_final = (float*)d_out + NB;            // [50000,128] x

  float* buf0 = (float*)d_ws;                     // x entering a layer
  float* buf1 = buf0 + NB;                        // xn (normalized x)
  float* buf2 = buf1 + NB;                        // ping-pong accumulator

  const dim3 gemm_grid(N_NODES / 16), gemm_blk(256);
  const dim3 norm_grid(N_NODES),      norm_blk(128);
  const dim3 edge_grid((N_EDGES + 7) / 8), edge_blk(256);
  const size_t NBYTES = NB * sizeof(float);

  // x = feat @ lin_w + lin_b
  gemm_bias_wmma<<<gemm_grid, gemm_blk, 0, stream>>>(feat, lin_w, lin_b, buf0);

  // ---- layer 1 ----
  channel_norm<<<norm_grid, norm_blk, 0, stream>>>(buf0, buf1);        // xn
  hipMemcpyAsync(buf0, buf1, NBYTES, hipMemcpyDeviceToDevice, stream); // acc=xn
  route_edges<<<edge_grid, edge_blk, 0, stream>>>(buf1, buf1, buf0, src, trg);
  channel_norm<<<norm_grid, norm_blk, 0, stream>>>(buf0, buf0);        // c=norm
  hipMemcpyAsync(buf2, buf1, NBYTES, hipMemcpyDeviceToDevice, stream);
  route_edges<<<edge_grid, edge_blk, 0, stream>>>(buf1, buf0, buf2, src, trg);
  channel_norm<<<norm_grid, norm_blk, 0, stream>>>(buf2, buf2);
  hipMemcpyAsync(buf0, buf1, NBYTES, hipMemcpyDeviceToDevice, stream);
  route_edges<<<edge_grid, edge_blk, 0, stream>>>(buf1, buf2, buf0, src, trg);
  // layer-1 output in buf0 (no final normalize)

  // ---- layer 2 ----
  channel_norm<<<norm_grid, norm_blk, 0, stream>>>(buf0, buf1);        // xn
  hipMemcpyAsync(buf2, buf1, NBYTES, hipMemcpyDeviceToDevice, stream);
  route_edges<<<edge_grid, edge_blk, 0, stream>>>(buf1, buf1, buf2, src, trg);
  channel_norm<<<norm_grid, norm_blk, 0, stream>>>(buf2, buf2);
  hipMemcpyAsync(buf0, buf1, NBYTES, hipMemcpyDeviceToDevice, stream);
  route_edges<<<edge_grid, edge_blk, 0, stream>>>(buf1, buf2, buf0, src, trg);
  channel_norm<<<norm_grid, norm_blk, 0, stream>>>(buf0, buf0);
  hipMemcpyAsync(x_final, buf1, NBYTES, hipMemcpyDeviceToDevice, stream);
  route_edges<<<edge_grid, edge_blk, 0, stream>>>(buf1, buf0, x_final, src, trg);
  // x_final = final features (second half of d_out)

  // out = x_final @ mlp_w + mlp_b
  gemm_bias_wmma<<<gemm_grid, gemm_blk, 0, stream>>>(x_final, mlp_w, mlp_b, out_buf);
}